// DiscreteLinear_36575941493096
// MI455X (gfx1250) — compile-verified
//
#include <hip/hip_runtime.h>

// ---------------------------------------------------------------------------
// DiscreteLinear: z[b,i] = sum_j W[a[b]][i][j] * x[b][j] + bias[a[b]][i]
// Grouped-GEMM: bucket rows by action, then per-expert fp32 WMMA tiles.
//   - W tiles staged global->LDS with GLOBAL_LOAD_ASYNC_TO_LDS_B128 (ASYNCcnt)
//   - A fragments streamed from global (b64/lane), B fragments from padded LDS
//   - V_WMMA_F32_16X16X4_F32 accumulation (exact fp32, matches reference)
// ---------------------------------------------------------------------------

#define NACT    64
#define DIM     512
#define BATCH   2048
#define NWAVES  4
#define WSTRIDE 516   // padded LDS row stride (floats) -> conflict-free ds_load_b64
#define DL_USE_ASYNC 1

typedef __attribute__((ext_vector_type(2))) float v2f;
typedef __attribute__((ext_vector_type(8))) float v8f;

__global__ void dl_init_counts(int* counts) {
    if (threadIdx.x < NACT) counts[threadIdx.x] = 0;
}

__global__ void dl_scatter(const int* __restrict__ act, int* __restrict__ counts,
                           int* __restrict__ perm) {
    int b = blockIdx.x * blockDim.x + threadIdx.x;
    if (b < BATCH) {
        int e = act[b];
        int pos = atomicAdd(&counts[e], 1);   // order-independent per-row results
        perm[e * BATCH + pos] = b;
    }
}

__global__ __launch_bounds__(NWAVES * 32)
void dl_grouped_gemm(const float* __restrict__ x,
                     const float* __restrict__ weight,
                     const float* __restrict__ bias,
                     const int*   __restrict__ perm,
                     const int*   __restrict__ counts,
                     float*       __restrict__ out) {
    extern __shared__ float Ws[];        // 64 rows x WSTRIDE floats (132 KB)

    const int e      = blockIdx.x;            // expert
    const int nblk   = blockIdx.y * 64;       // first output column of block
    const int tid    = threadIdx.x;
    const int lane   = tid & 31;
    const int wave   = tid >> 5;
    const int nbase  = nblk + wave * 16;      // this wave's N-tile
    const int cnt    = counts[e];

    // --- stage 64 W rows (nblk..nblk+63, all 512 cols) into LDS ------------
    {
        const float* wsrc = weight + ((size_t)e * DIM + nblk) * DIM;
#if DL_USE_ASYNC
        const unsigned lds_base = (unsigned)(size_t)(void*)Ws;
        for (int i = tid; i < 64 * (DIM / 4); i += NWAVES * 32) {
            int r  = i >> 7;                  // / (DIM/4)
            int c4 = i & 127;                 // % (DIM/4)
            unsigned long long gaddr =
                (unsigned long long)(const void*)(wsrc + (size_t)r * DIM + c4 * 4);
            unsigned laddr = lds_base + (unsigned)(r * WSTRIDE + c4 * 4) * 4u;
            // async global -> LDS, 16B per lane, tracked by ASYNCcnt
            asm volatile("global_load_async_to_lds_b128 %0, %1, off"
                         :: "v"(laddr), "v"(gaddr) : "memory");
        }
        asm volatile("s_wait_asynccnt 0x0" ::: "memory");
#else
        for (int i = tid; i < 64 * (DIM / 4); i += NWAVES * 32) {
            int r  = i >> 7;
            int c4 = i & 127;
            float4 v = ((const float4*)(wsrc + (size_t)r * DIM))[c4];
            *(float4*)(&Ws[r * WSTRIDE + c4 * 4]) = v;
        }
#endif
    }
    __syncthreads();

    if (cnt == 0) return;   // uniform per block

    // fragment lane mapping (ISA 16x4 f32 A layout; B mirrored)
    const int m    = lane & 15;           // A row / B col / bias col
    const int koff = (lane >> 4) << 1;    // {0,2}: K pair handled by this half
    const float* wlds = &Ws[(wave * 16 + m) * WSTRIDE];

    const float bval = bias[(size_t)e * DIM + nbase + m];

    const int ntiles = (cnt + 15) >> 4;
    for (int t = 0; t < ntiles; ++t) {
        const int rbase = t * 16;
        const int ridx  = rbase + m;
        const int rclmp = ridx < cnt ? ridx : cnt - 1;    // clamp tail loads
        const int row   = perm[e * BATCH + rclmp];
        const float* xrow = x + (size_t)row * DIM;

        // pull this lane's x row toward the caches ahead of the b64 stream
        __builtin_prefetch(xrow, 0, 0);
        __builtin_prefetch(xrow + 128, 0, 0);
        __builtin_prefetch(xrow + 256, 0, 0);
        __builtin_prefetch(xrow + 384, 0, 0);

        v8f c = {0.f, 0.f, 0.f, 0.f, 0.f, 0.f, 0.f, 0.f};

        #pragma unroll 8
        for (int k0 = 0; k0 < DIM; k0 += 4) {
            // A fragment: x[row][k0+koff .. +1]  (global_load_b64, streamed)
            const v2f a = *(const v2f*)(xrow + k0 + koff);
            // B fragment: W[e][nbase+m][k0+koff .. +1] from LDS (ds_load_b64)
            const v2f b = *(const v2f*)(wlds + k0 + koff);
            c = __builtin_amdgcn_wmma_f32_16x16x4_f32(
                    /*neg_a=*/false, a, /*neg_b=*/false, b,
                    /*c_mod=*/(short)0, c, /*reuse_a=*/false, /*reuse_b=*/false);
        }

        // C/D layout: VGPR r -> M=r (lanes 0-15), M=r+8 (lanes 16-31); N=lane&15
        const int mrow = (lane >> 4) * 8;
        #pragma unroll
        for (int r = 0; r < 8; ++r) {
            int rr = rbase + mrow + r;
            int orow = __shfl(row, mrow + r, 32);   // row index lives in lanes 0-15
            if (rr < cnt) {
                out[(size_t)orow * DIM + nbase + m] = c[r] + bval;
            }
        }
    }
}

extern "C" void kernel_launch(void* const* d_in, const int* in_sizes, int n_in,
                              void* d_out, int out_size, void* d_ws, size_t ws_size,
                              hipStream_t stream) {
    const float* x      = (const float*)d_in[0];
    const int*   act    = (const int*)  d_in[1];
    const float* weight = (const float*)d_in[2];
    const float* bias   = (const float*)d_in[3];
    float*       out    = (float*)d_out;

    int* counts = (int*)d_ws;            // 64 ints
    int* perm   = counts + 64;           // 64 * 2048 ints

    dl_init_counts<<<1, 64, 0, stream>>>(counts);
    dl_scatter<<<(BATCH + 255) / 256, 256, 0, stream>>>(act, counts, perm);

    dim3 grid(NACT, DIM / 64);           // 64 experts x 8 blocks of 64 columns
    size_t lds_bytes = (size_t)64 * WSTRIDE * sizeof(float);   // 132 KB
    dl_grouped_gemm<<<grid, NWAVES * 32, lds_bytes, stream>>>(x, weight, bias,
                                                              perm, counts, out);
}